// KnnTransformerNetwork_35347580846883
// MI455X (gfx1250) — compile-verified
//
#include <hip/hip_runtime.h>
#include <math.h>

#define M_PTS 16384
#define DM    256
#define NHEAD 4
#define DHEAD 64
#define KNN   16
#define DFFN  1024
#define LN_EPS 1e-5f

typedef float v2f __attribute__((ext_vector_type(2)));
typedef float v8f __attribute__((ext_vector_type(8)));

// ---------------------------------------------------------------------------
// Wave32 reductions
// ---------------------------------------------------------------------------
__device__ __forceinline__ float wave32_sum(float v) {
    v += __shfl_xor(v, 1, 32);
    v += __shfl_xor(v, 2, 32);
    v += __shfl_xor(v, 4, 32);
    v += __shfl_xor(v, 8, 32);
    v += __shfl_xor(v, 16, 32);
    return v;
}
__device__ __forceinline__ float group8_sum(float v) {
    v += __shfl_xor(v, 1, 32);
    v += __shfl_xor(v, 2, 32);
    v += __shfl_xor(v, 4, 32);
    return v;
}

// ---------------------------------------------------------------------------
// LayerNorm: one wave per row of 256, 8 floats / lane
// ---------------------------------------------------------------------------
__global__ __launch_bounds__(256)
void ln_kernel(const float* __restrict__ x, const float* __restrict__ g,
               const float* __restrict__ b, float* __restrict__ out) {
    const int wave = threadIdx.x >> 5;
    const int lane = threadIdx.x & 31;
    const int row  = blockIdx.x * 8 + wave;
    const int j0   = lane * 8;

    const float* xr = x + (size_t)row * DM + j0;
    float4 p0 = *(const float4*)(xr);
    float4 p1 = *(const float4*)(xr + 4);
    float v[8] = {p0.x, p0.y, p0.z, p0.w, p1.x, p1.y, p1.z, p1.w};

    float s = 0.f;
#pragma unroll
    for (int i = 0; i < 8; ++i) s += v[i];
    const float mean = wave32_sum(s) * (1.0f / DM);

    float q = 0.f;
#pragma unroll
    for (int i = 0; i < 8; ++i) { float d = v[i] - mean; q += d * d; }
    const float var = wave32_sum(q) * (1.0f / DM);
    const float inv = 1.0f / sqrtf(var + LN_EPS);

    float4 g0 = *(const float4*)(g + j0);
    float4 g1 = *(const float4*)(g + j0 + 4);
    float4 b0 = *(const float4*)(b + j0);
    float4 b1 = *(const float4*)(b + j0 + 4);
    float gg[8] = {g0.x, g0.y, g0.z, g0.w, g1.x, g1.y, g1.z, g1.w};
    float bb[8] = {b0.x, b0.y, b0.z, b0.w, b1.x, b1.y, b1.z, b1.w};

    float* orow = out + (size_t)row * DM + j0;
    float o[8];
#pragma unroll
    for (int i = 0; i < 8; ++i) o[i] = (v[i] - mean) * inv * gg[i] + bb[i];
    *(float4*)(orow)     = make_float4(o[0], o[1], o[2], o[3]);
    *(float4*)(orow + 4) = make_float4(o[4], o[5], o[6], o[7]);
}

// ---------------------------------------------------------------------------
// WMMA fp32 GEMM: C[M,N] = epilogue(A[M,K] @ B[K,N])
// Block = 256 threads (8 waves); block tile = 256 x 64; wave tile = 32 x 64.
// B panel (64 x 64) staged through LDS in K-pair-interleaved layout:
//   lds_b[p*BPSTRIDE + 2*c + parity] = B[k0 + 2p + parity][n_base + c]
// so a WMMA B fragment is ONE aligned ds_load_b64, and BPSTRIDE=160
// (stride mod 64 banks == 32) puts the two half-waves (rows p / p+1) on
// disjoint bank halves -> conflict-free.
// MODE: 0 = plain, 1 = +residual, 2 = bias+GELU(tanh), 3 = bias+residual
// ---------------------------------------------------------------------------
#define BK 64
#define BPSTRIDE 160  // floats per K-pair row (128 data + 32 pad)
template <int MODE>
__global__ __launch_bounds__(256)
void gemm_kernel(const float* __restrict__ A, const float* __restrict__ B,
                 const float* __restrict__ bias, const float* __restrict__ res,
                 float* __restrict__ C, int Kdim, int N) {
    __shared__ float lds_b[(BK / 2) * BPSTRIDE];

    const int tid  = threadIdx.x;
    const int wave = tid >> 5;
    const int lane = tid & 31;
    const int m_base = blockIdx.x * 256 + wave * 32;
    const int n_base = blockIdx.y * 64;
    const int r     = lane & 15;       // row-in-tile (A) / col-in-tile (B)
    const int half  = lane >> 4;       // 0 or 1
    const int koff  = half * 2;        // which K pair this half-wave owns

    v8f acc[2][4];
#pragma unroll
    for (int mt = 0; mt < 2; ++mt)
#pragma unroll
        for (int t = 0; t < 4; ++t)
#pragma unroll
            for (int i = 0; i < 8; ++i) acc[mt][t][i] = 0.0f;

    const float* arow0 = A + (size_t)(m_base + r) * Kdim;
    const float* arow1 = arow0 + (size_t)16 * Kdim;
    // LDS read base for this lane: pair row advanced by `half`, column r of
    // each of the 4 tiles (immediate offsets handle kk and tile).
    const float* lb = lds_b + half * BPSTRIDE + 2 * r;

    for (int k0 = 0; k0 < Kdim; k0 += BK) {
        __syncthreads();
        // Stage B panel: each thread handles 8 K-pair units (p, c).
#pragma unroll
        for (int i = 0; i < (BK / 2) * 64 / 256; ++i) {  // 8 iterations
            int u = i * 256 + tid;
            int p = u >> 6, c = u & 63;
            const float* bp = B + (size_t)(k0 + 2 * p) * N + n_base + c;
            float2 w;
            w.x = bp[0];
            w.y = bp[N];
            *(float2*)(lds_b + p * BPSTRIDE + 2 * c) = w;
        }
        __syncthreads();

        __builtin_prefetch(arow0 + k0 + BK, 0, 3);
        __builtin_prefetch(arow1 + k0 + BK, 0, 3);

#pragma unroll
        for (int kk = 0; kk < BK; kk += 4) {
            const int ka = k0 + kk + koff;          // even -> 8B aligned
            v2f a0 = *(const v2f*)(arow0 + ka);
            v2f a1 = *(const v2f*)(arow1 + ka);
            const float* lrow = lb + (kk >> 1) * BPSTRIDE;
#pragma unroll
            for (int t = 0; t < 4; ++t) {
                v2f bf = *(const v2f*)(lrow + t * 32);  // one ds_load_b64
                acc[0][t] = __builtin_amdgcn_wmma_f32_16x16x4_f32(
                    false, a0, false, bf, (short)0, acc[0][t], false, false);
                acc[1][t] = __builtin_amdgcn_wmma_f32_16x16x4_f32(
                    false, a1, false, bf, (short)0, acc[1][t], false, false);
            }
        }
    }

    const int rowBase = m_base + half * 8;  // C: vgpr i -> row i or i+8
#pragma unroll
    for (int mt = 0; mt < 2; ++mt) {
#pragma unroll
        for (int t = 0; t < 4; ++t) {
            const int col = n_base + t * 16 + r;
            const float bv = (MODE == 2 || MODE == 3) ? bias[col] : 0.0f;
#pragma unroll
            for (int i = 0; i < 8; ++i) {
                const int row = rowBase + mt * 16 + i;
                float v = acc[mt][t][i];
                if (MODE == 2 || MODE == 3) v += bv;
                if (MODE == 2) {  // tanh-approx GELU (jax.nn.gelu default)
                    float u = v;
                    float c = 0.7978845608028654f * (u + 0.044715f * u * u * u);
                    v = 0.5f * u * (1.0f + tanhf(c));
                }
                if (MODE == 1 || MODE == 3) v += res[(size_t)row * N + col];
                C[(size_t)row * N + col] = v;
            }
        }
    }
}

// ---------------------------------------------------------------------------
// Gather attention: one wave per query point. Lane l owns dims [8l, 8l+8);
// 8-lane group g = head g (dims 64g..64g+63). Online softmax over K=16.
// ---------------------------------------------------------------------------
__global__ __launch_bounds__(256)
void attn_kernel(const float* __restrict__ q, const float* __restrict__ kall,
                 const float* __restrict__ vall, const float* __restrict__ qcoord,
                 const float* __restrict__ ccoord, const int* __restrict__ knn,
                 const float* __restrict__ pe_w, const float* __restrict__ pe_b,
                 float* __restrict__ out) {
    const int wave = threadIdx.x >> 5;
    const int lane = threadIdx.x & 31;
    const int m    = blockIdx.x * 8 + wave;
    const int j0   = lane * 8;

    float qv[8], pw0[8], pw1[8], pw2[8], pb[8];
    {
        float4 a0 = *(const float4*)(q + (size_t)m * DM + j0);
        float4 a1 = *(const float4*)(q + (size_t)m * DM + j0 + 4);
        qv[0]=a0.x; qv[1]=a0.y; qv[2]=a0.z; qv[3]=a0.w;
        qv[4]=a1.x; qv[5]=a1.y; qv[6]=a1.z; qv[7]=a1.w;
#pragma unroll
        for (int i = 0; i < 8; ++i) {
            pw0[i] = pe_w[j0 + i];
            pw1[i] = pe_w[DM + j0 + i];
            pw2[i] = pe_w[2 * DM + j0 + i];
            pb[i]  = pe_b[j0 + i];
        }
    }
    const float cx = qcoord[m * 3 + 0];
    const float cy = qcoord[m * 3 + 1];
    const float cz = qcoord[m * 3 + 2];

    float m_run = -INFINITY, l_run = 0.0f;
    float o[8];
#pragma unroll
    for (int i = 0; i < 8; ++i) o[i] = 0.0f;

    for (int kn = 0; kn < KNN; ++kn) {
        const int idx    = knn[(size_t)m * KNN + kn];
        const bool valid = idx >= 0;
        const int ic     = valid ? idx : 0;

        const float rx = cx - ccoord[ic * 3 + 0];
        const float ry = cy - ccoord[ic * 3 + 1];
        const float rz = cz - ccoord[ic * 3 + 2];

        const float* kr = kall + (size_t)ic * DM + j0;
        const float* vr = vall + (size_t)ic * DM + j0;
        float4 k0_ = *(const float4*)(kr);
        float4 k1_ = *(const float4*)(kr + 4);
        float4 v0_ = *(const float4*)(vr);
        float4 v1_ = *(const float4*)(vr + 4);
        float kf[8] = {k0_.x, k0_.y, k0_.z, k0_.w, k1_.x, k1_.y, k1_.z, k1_.w};
        float vf[8] = {v0_.x, v0_.y, v0_.z, v0_.w, v1_.x, v1_.y, v1_.z, v1_.w};

        float s = 0.0f;
        float vg[8];
#pragma unroll
        for (int i = 0; i < 8; ++i) {
            float pe = fmaf(rx, pw0[i], fmaf(ry, pw1[i], fmaf(rz, pw2[i], pb[i])));
            float kg = kf[i] + pe;
            vg[i] = vf[i] + pe;
            s = fmaf(qv[i], kg, s);
        }
        s = group8_sum(s) * 0.125f;  // 1/sqrt(64)
        if (!valid) s = -1e9f;

        const float m_new = fmaxf(m_run, s);
        const float alpha = expf(m_run - m_new);
        const float p     = expf(s - m_new);
        l_run = l_run * alpha + p;
#pragma unroll
        for (int i = 0; i < 8; ++i) o[i] = o[i] * alpha + p * vg[i];
        m_run = m_new;
    }

    const float inv = 1.0f / l_run;
    float* orow = out + (size_t)m * DM + j0;
    *(float4*)(orow)     = make_float4(o[0]*inv, o[1]*inv, o[2]*inv, o[3]*inv);
    *(float4*)(orow + 4) = make_float4(o[4]*inv, o[5]*inv, o[6]*inv, o[7]*inv);
}

// ---------------------------------------------------------------------------
// Orchestration
// ---------------------------------------------------------------------------
extern "C" void kernel_launch(void* const* d_in, const int* in_sizes, int n_in,
                              void* d_out, int out_size, void* d_ws, size_t ws_size,
                              hipStream_t stream) {
    const float* feat_a  = (const float*)d_in[0];
    const float* coord_a = (const float*)d_in[1];
    const float* feat_b  = (const float*)d_in[2];
    const float* coord_b = (const float*)d_in[3];
    const float* Wq      = (const float*)d_in[4];
    const float* Wk      = (const float*)d_in[5];
    const float* Wv      = (const float*)d_in[6];
    const float* Wo      = (const float*)d_in[7];
    const float* ln1_g   = (const float*)d_in[8];
    const float* ln1_b   = (const float*)d_in[9];
    const float* pe_w    = (const float*)d_in[10];
    const float* pe_b    = (const float*)d_in[11];
    const float* ffn_w1  = (const float*)d_in[12];
    const float* ffn_b1  = (const float*)d_in[13];
    const float* ffn_w2  = (const float*)d_in[14];
    const float* ffn_b2  = (const float*)d_in[15];
    const float* ln2_g   = (const float*)d_in[16];
    const float* ln2_b   = (const float*)d_in[17];
    const int*   knn_a2a = (const int*)d_in[18];
    const int*   knn_a2b = (const int*)d_in[19];

    float* x = (float*)d_out;  // evolving features [M, 256]

    // Workspace layout (floats): h[M*DFF] | xn[M*D] | q[M*D] | k[M*D] | v[M*D]
    float* buf_h  = (float*)d_ws;
    float* buf_xn = buf_h  + (size_t)M_PTS * DFFN;
    float* buf_q  = buf_xn + (size_t)M_PTS * DM;
    float* buf_k  = buf_q  + (size_t)M_PTS * DM;
    float* buf_v  = buf_k  + (size_t)M_PTS * DM;
    float* buf_at = buf_xn;  // xn dead once q is computed; reuse for attn output

    hipMemcpyAsync(x, feat_a, sizeof(float) * (size_t)M_PTS * DM,
                   hipMemcpyDeviceToDevice, stream);

    const dim3 blk256(256);
    const dim3 gRow(M_PTS / 8);
    const dim3 gGemmD(M_PTS / 256, DM / 64);
    const dim3 gGemmF(M_PTS / 256, DFFN / 64);

    for (int b = 0; b < 2; ++b) {
        const float* ctx  = (b == 0) ? feat_a  : feat_b;
        const float* ctxc = (b == 0) ? coord_a : coord_b;
        const int*   knn  = (b == 0) ? knn_a2a : knn_a2b;
        const size_t wOff = (size_t)b * DM * DM;

        // xn = LN1(x)
        ln_kernel<<<gRow, blk256, 0, stream>>>(x, ln1_g + b * DM, ln1_b + b * DM, buf_xn);
        // q = xn @ Wq ; k_all = ctx @ Wk ; v_all = ctx @ Wv   (ctx is NOT normed)
        gemm_kernel<0><<<gGemmD, blk256, 0, stream>>>(buf_xn, Wq + wOff, nullptr, nullptr, buf_q, DM, DM);
        gemm_kernel<0><<<gGemmD, blk256, 0, stream>>>(ctx,    Wk + wOff, nullptr, nullptr, buf_k, DM, DM);
        gemm_kernel<0><<<gGemmD, blk256, 0, stream>>>(ctx,    Wv + wOff, nullptr, nullptr, buf_v, DM, DM);
        // gather attention with relative PE + online softmax
        attn_kernel<<<gRow, blk256, 0, stream>>>(buf_q, buf_k, buf_v, coord_a, ctxc, knn,
                                                 pe_w + (size_t)b * 3 * DM, pe_b + b * DM, buf_at);
        // x = x + attn @ Wo
        gemm_kernel<1><<<gGemmD, blk256, 0, stream>>>(buf_at, Wo + wOff, nullptr, x, x, DM, DM);
        // xn = LN2(x)
        ln_kernel<<<gRow, blk256, 0, stream>>>(x, ln2_g + b * DM, ln2_b + b * DM, buf_xn);
        // h = gelu(xn @ W1 + b1)
        gemm_kernel<2><<<gGemmF, blk256, 0, stream>>>(buf_xn, ffn_w1 + (size_t)b * DM * DFFN,
                                                      ffn_b1 + b * DFFN, nullptr, buf_h, DM, DFFN);
        // x = x + (h @ W2 + b2)
        gemm_kernel<3><<<gGemmD, blk256, 0, stream>>>(buf_h, ffn_w2 + (size_t)b * DFFN * DM,
                                                      ffn_b2 + b * DM, x, x, DFFN, DM);
    }
    (void)in_sizes; (void)n_in; (void)out_size; (void)ws_size;
}